// InterpreterUnit_23725399343445
// MI455X (gfx1250) — compile-verified
//
#include <hip/hip_runtime.h>

// Elementwise "interpreter" over N rows of D=128 f32.
// Bandwidth-bound: one wave32 per row (32 lanes x float4 = 128 floats),
// wave-uniform scalar branching to skip dead loads, NT-hinted B128 vector
// memory ops, and gfx1250 global_prefetch for the next row.

typedef __attribute__((ext_vector_type(4))) float v4f;

#define WAVES_PER_BLOCK 8
#define ROWS_PER_WAVE   4
#define DIM             128
#define IN_STRIDE       131   // 3 flags + 128 payload floats per inputs row

// Broadcast a per-lane (but actually uniform) float into an SGPR so the
// compiler emits scalar compares + s_cbranch (divergence-free branch skip).
__device__ __forceinline__ float uniform_f(float x) {
    return __builtin_bit_cast(float,
        __builtin_amdgcn_readfirstlane(__builtin_bit_cast(int, x)));
}

__global__ void __launch_bounds__(WAVES_PER_BLOCK * 32)
interp_rowwave_kernel(const int*   __restrict__ arities,
                      const float* __restrict__ inputs,     // [n][131]
                      const float* __restrict__ children,   // [2][n][128]
                      float*       __restrict__ out,        // [n][128]
                      int n)
{
    const int lane = threadIdx.x & 31;
    const int wave = __builtin_amdgcn_readfirstlane((int)(threadIdx.x >> 5));
    const long long row0 =
        ((long long)blockIdx.x * WAVES_PER_BLOCK + wave) * ROWS_PER_WAVE;

    const float* __restrict__ c0b = children;                       // children[0]
    const float* __restrict__ c1b = children + (long long)n * DIM;  // children[1]

#pragma unroll
    for (int i = 0; i < ROWS_PER_WAVE; ++i) {
        const long long row = row0 + i;
        if (row >= n) return;  // wave-uniform exit

        // gfx1250 global_prefetch_b8: pull next row's children lines while
        // this row computes (512B per buffer, per wave).
        if (i + 1 < ROWS_PER_WAVE && row + 1 < n) {
            const long long poff = (row + 1) * DIM + (long long)lane * 4;
            __builtin_prefetch(c0b + poff, 0, 0);
            __builtin_prefetch(c1b + poff, 0, 0);
        }

        const float* __restrict__ inrow = inputs + row * (long long)IN_STRIDE;
        // Flags: uniform address -> one coalesced fetch; force SGPR compares.
        const float f0 = uniform_f(inrow[0]);
        const float f1 = uniform_f(inrow[1]);
        const float f2 = uniform_f(inrow[2]);

        const long long coff = row * DIM + (long long)lane * 4;  // 16B aligned
        v4f r;

        if (f0 > 0.0f) {
            // add branch: c0 + c1
            v4f a = __builtin_nontemporal_load((const v4f*)(c0b + coff));
            v4f b = __builtin_nontemporal_load((const v4f*)(c1b + coff));
            r = a + b;
        } else if (f1 > 0.0f) {
            // mul branch: c0 * c1
            v4f a = __builtin_nontemporal_load((const v4f*)(c0b + coff));
            v4f b = __builtin_nontemporal_load((const v4f*)(c1b + coff));
            r = a * b;
        } else if (f2 > 0.0f) {
            // neg / sub branch (arity only read here, scalar)
            const int ar = __builtin_amdgcn_readfirstlane(arities[row]);
            v4f a = __builtin_nontemporal_load((const v4f*)(c0b + coff));
            if (ar == 1) {
                r = -a;
            } else {
                v4f b = __builtin_nontemporal_load((const v4f*)(c1b + coff));
                r = a - b;
            }
        } else {
            // passthrough: inputs[:, 3:]  (row stride 524B -> only 4B aligned,
            // use scalar NT loads; this branch runs ~12.5% of rows)
            const float* __restrict__ t = inrow + 3 + lane * 4;
            r.x = __builtin_nontemporal_load(t + 0);
            r.y = __builtin_nontemporal_load(t + 1);
            r.z = __builtin_nontemporal_load(t + 2);
            r.w = __builtin_nontemporal_load(t + 3);
        }

        __builtin_nontemporal_store(r, (v4f*)(out + coff));
    }
}

extern "C" void kernel_launch(void* const* d_in, const int* in_sizes, int n_in,
                              void* d_out, int out_size, void* d_ws, size_t ws_size,
                              hipStream_t stream) {
    const int*   arities  = (const int*)d_in[0];
    const float* inputs   = (const float*)d_in[1];   // [n][131]
    const float* children = (const float*)d_in[2];   // [2][n][128]
    float*       out      = (float*)d_out;           // [n][128]
    const int n = in_sizes[0];

    const long long waves  = ((long long)n + ROWS_PER_WAVE - 1) / ROWS_PER_WAVE;
    const int       blocks = (int)((waves + WAVES_PER_BLOCK - 1) / WAVES_PER_BLOCK);

    interp_rowwave_kernel<<<blocks, WAVES_PER_BLOCK * 32, 0, stream>>>(
        arities, inputs, children, out, n);
}